// GAT_61186104099457
// MI455X (gfx1250) — compile-verified
//
#include <hip/hip_runtime.h>
#include <hip/hip_bf16.h>
#include <math.h>

typedef float v2f __attribute__((ext_vector_type(2)));
typedef float v8f __attribute__((ext_vector_type(8)));
typedef int   v4i __attribute__((ext_vector_type(4)));

#define DIN 512
#define HF  256
#define NH  8
#define CC  40
#define NEG_SLOPE 0.2f
#define SEPS 1e-16f
#define LDA 516   // padded LDS stride (floats) -> 2064B row stride (129*16B), spreads banks

#define AS1 __attribute__((address_space(1)))
#define AS3 __attribute__((address_space(3)))

#if defined(__has_builtin)
#if __has_builtin(__builtin_amdgcn_global_load_async_to_lds_b128)
#define HAVE_ASYNC_LDS 1
#endif
#endif

// ---- order-preserving float<->uint encoding for atomicMax on floats ----
__device__ __forceinline__ unsigned fenc(float f) {
  unsigned u = __float_as_uint(f);
  return (u & 0x80000000u) ? ~u : (u | 0x80000000u);
}
__device__ __forceinline__ float fdec(unsigned e) {
  return (e & 0x80000000u) ? __uint_as_float(e & 0x7FFFFFFFu)
                           : __uint_as_float(~e);
}
__device__ __forceinline__ float lrelu(float v) {
  return v > 0.0f ? v : NEG_SLOPE * v;
}

// ---------------------------------------------------------------- fill
__global__ void fill_zero_k(float* __restrict__ p, long long n) {
  long long i = (long long)blockIdx.x * blockDim.x + threadIdx.x;
  if (i < n) p[i] = 0.0f;
}

// ------------------------------------------------- GEMM1: h1 = x @ W1
// Block: 256 thr = 8 waves. blockIdx.x = 16-row M tile, blockIdx.y in {0,1}
// selects 128-col group; wave w handles 16 cols. A tile staged in LDS via
// async-to-LDS DMA (ASYNCcnt) when available.
__global__ __launch_bounds__(256) void gemm1_wmma(
    const float* __restrict__ x, const float* __restrict__ W1,
    float* __restrict__ h1, int N) {
  __shared__ __align__(16) float As[16][LDA];
  const int tid = threadIdx.x;
  int m0 = blockIdx.x * 16;
  if (m0 + 16 > N) m0 = N - 16;   // clamp: overlapping rows recompute identically

  // cooperative A-tile load: 16 x 512 floats = 2048 float4
  for (int i = tid; i < 16 * 128; i += 256) {
    int r  = i >> 7;
    int c4 = i & 127;
    const float4* gsrc = (const float4*)(x + (size_t)(m0 + r) * DIN) + c4;
    float* ldst = &As[r][c4 * 4];
#ifdef HAVE_ASYNC_LDS
    __builtin_amdgcn_global_load_async_to_lds_b128(
        (AS1 v4i*)(uintptr_t)gsrc,
        (AS3 v4i*)(unsigned)(uintptr_t)ldst, 0, 0);
#else
    float4 v = *gsrc;
    ldst[0] = v.x; ldst[1] = v.y; ldst[2] = v.z; ldst[3] = v.w;
#endif
  }
#ifdef HAVE_ASYNC_LDS
#if __has_builtin(__builtin_amdgcn_s_wait_asynccnt)
  __builtin_amdgcn_s_wait_asynccnt(0);
#else
  asm volatile("s_wait_asynccnt 0x0" ::: "memory");
#endif
#endif
  __syncthreads();

  const int wave  = tid >> 5;
  const int lane  = tid & 31;
  const int c0    = blockIdx.y * 128 + wave * 16;
  const int nrow  = lane & 15;        // M for A frag / N for B frag
  const int khalf = lane >> 4;        // 0: K 0..1, 1: K 2..3

  v8f acc = {};
#pragma unroll 4
  for (int k0 = 0; k0 < DIN; k0 += 4) {
    v2f a, b;
    a.x = As[nrow][k0 + khalf * 2 + 0];
    a.y = As[nrow][k0 + khalf * 2 + 1];
    const float* bp = W1 + (size_t)(k0 + khalf * 2) * HF + c0 + nrow;
    b.x = bp[0];
    b.y = bp[HF];
    acc = __builtin_amdgcn_wmma_f32_16x16x4_f32(
        false, a, false, b, (short)0, acc, false, false);
  }
  // D layout: VGPR i -> row m0 + khalf*8 + i, col c0 + nrow
  float* op = h1 + (size_t)(m0 + khalf * 8) * HF + c0 + nrow;
#pragma unroll
  for (int i = 0; i < 8; i++) op[(size_t)i * HF] = acc[i];
}

// ------------------------------------------------- GEMM2: z2 = h2 @ W2
// 16x16 tile per wave; col tiles 0..2 cover 40 cols (guarded).
__global__ __launch_bounds__(256) void gemm2_wmma(
    const float* __restrict__ h2, const float* __restrict__ W2,
    float* __restrict__ z2, int N, int nWaves) {
  const int wid = blockIdx.x * 8 + (threadIdx.x >> 5);
  if (wid >= nWaves) return;
  const int mt = wid / 3, ct = wid % 3;
  int m0 = mt * 16;
  if (m0 + 16 > N) m0 = N - 16;   // clamp: duplicate rows write identical values
  const int c0 = ct * 16;
  const int lane  = threadIdx.x & 31;
  const int nrow  = lane & 15;
  const int khalf = lane >> 4;
  const int col   = c0 + nrow;

  v8f acc = {};
#pragma unroll 4
  for (int k0 = 0; k0 < HF; k0 += 4) {
    int k = k0 + khalf * 2;
    v2f a, b;
    const float* ap = h2 + (size_t)(m0 + nrow) * HF + k;
    a.x = ap[0];
    a.y = ap[1];
    b.x = (col < CC) ? W2[(size_t)k * CC + col] : 0.0f;
    b.y = (col < CC) ? W2[(size_t)(k + 1) * CC + col] : 0.0f;
    acc = __builtin_amdgcn_wmma_f32_16x16x4_f32(
        false, a, false, b, (short)0, acc, false, false);
  }
  if (col < CC) {
#pragma unroll
    for (int i = 0; i < 8; i++)
      z2[(size_t)(m0 + khalf * 8 + i) * CC + col] = acc[i];
  }
}

// ------------------------------------------- attention projections, L1
__global__ void att1_k(const float* __restrict__ h1,
                       const float* __restrict__ attS,
                       const float* __restrict__ attD,
                       float* __restrict__ aS, float* __restrict__ aD, int N) {
  int i = blockIdx.x * blockDim.x + threadIdx.x;
  if (i >= N * NH) return;
  int n = i >> 3, hd = i & 7;
  const float* hp = h1 + (size_t)n * HF + hd * 32;
  const float* as = attS + hd * 32;
  const float* ad = attD + hd * 32;
  float s = 0.0f, d = 0.0f;
#pragma unroll
  for (int f = 0; f < 32; f++) { float v = hp[f]; s += v * as[f]; d += v * ad[f]; }
  aS[i] = s; aD[i] = d;
}

// ------------------------------------------- attention projections, L2
__global__ void att2_k(const float* __restrict__ z2,
                       const float* __restrict__ attS,
                       const float* __restrict__ attD,
                       float* __restrict__ aS, float* __restrict__ aD, int N) {
  int n = blockIdx.x * blockDim.x + threadIdx.x;
  if (n >= N) return;
  const float* zp = z2 + (size_t)n * CC;
  float s = 0.0f, d = 0.0f;
#pragma unroll
  for (int c = 0; c < CC; c++) { float v = zp[c]; s += v * attS[c]; d += v * attD[c]; }
  aS[n] = s; aD[n] = d;
}

__device__ __forceinline__ void edge_sd(const long long* __restrict__ ei,
                                        int e, int E, int& src, int& dst) {
  if (e < E) { src = (int)ei[e]; dst = (int)ei[E + e]; }
  else       { src = e - E; dst = e - E; }
}

// ------------------------------------------- layer-1 edge phases (H=8)
__global__ void edge_max1(const long long* __restrict__ ei,
                          const float* __restrict__ aS, const float* __restrict__ aD,
                          unsigned* __restrict__ mx, int E, int N) {
  int i = blockIdx.x * blockDim.x + threadIdx.x;
  if (i >= (E + N) * NH) return;
  int e = i >> 3, hd = i & 7, src, dst;
  edge_sd(ei, e, E, src, dst);
  float v = lrelu(aS[src * NH + hd] + aD[dst * NH + hd]);
  atomicMax(&mx[dst * NH + hd], fenc(v));
}

__global__ void edge_sum1(const long long* __restrict__ ei,
                          const float* __restrict__ aS, const float* __restrict__ aD,
                          const unsigned* __restrict__ mx,
                          float* __restrict__ ex, float* __restrict__ s,
                          int E, int N) {
  int i = blockIdx.x * blockDim.x + threadIdx.x;
  if (i >= (E + N) * NH) return;
  int e = i >> 3, hd = i & 7, src, dst;
  edge_sd(ei, e, E, src, dst);
  float v = lrelu(aS[src * NH + hd] + aD[dst * NH + hd]);
  float m = fdec(mx[dst * NH + hd]);
  float x = expf(v - m);
  ex[i] = x;
  atomicAdd(&s[dst * NH + hd], x);
}

__global__ __launch_bounds__(256) void edge_aggr1(
    const long long* __restrict__ ei, const float* __restrict__ ex,
    const float* __restrict__ s, const float* __restrict__ h1,
    float* __restrict__ acc, int E, int N) {
  int t = threadIdx.x;          // t = hd*32 + f, 0..255
  int hd = t >> 5;
  int e0 = blockIdx.x * 4;
  int e1 = min(e0 + 4, E + N);
  int src, dst;
  edge_sd(ei, e0, E, src, dst);
  __builtin_prefetch(&h1[(size_t)src * HF + t], 0, 1);
  for (int e = e0; e < e1; e++) {
    int csrc = src, cdst = dst;
    if (e + 1 < e1) {           // prefetch next edge's source row
      edge_sd(ei, e + 1, E, src, dst);
      __builtin_prefetch(&h1[(size_t)src * HF + t], 0, 1);
    }
    float alpha = ex[(size_t)e * NH + hd] / (s[cdst * NH + hd] + SEPS);
    atomicAdd(&acc[(size_t)cdst * HF + t], alpha * h1[(size_t)csrc * HF + t]);
  }
}

// ------------------------------------------- bias + ELU (in place)
__global__ void elu_bias_k(float* __restrict__ a, const float* __restrict__ b,
                           long long total) {
  long long i = (long long)blockIdx.x * blockDim.x + threadIdx.x;
  if (i >= total) return;
  float v = a[i] + b[i & (HF - 1)];
  a[i] = v > 0.0f ? v : expm1f(v);
}

// ------------------------------------------- layer-2 edge phases (H=1)
__global__ void edge_max2(const long long* __restrict__ ei,
                          const float* __restrict__ aS, const float* __restrict__ aD,
                          unsigned* __restrict__ mx, int E, int N) {
  int e = blockIdx.x * blockDim.x + threadIdx.x;
  if (e >= E + N) return;
  int src, dst;
  edge_sd(ei, e, E, src, dst);
  atomicMax(&mx[dst], fenc(lrelu(aS[src] + aD[dst])));
}

__global__ void edge_sum2(const long long* __restrict__ ei,
                          const float* __restrict__ aS, const float* __restrict__ aD,
                          const unsigned* __restrict__ mx,
                          float* __restrict__ ex, float* __restrict__ s,
                          int E, int N) {
  int e = blockIdx.x * blockDim.x + threadIdx.x;
  if (e >= E + N) return;
  int src, dst;
  edge_sd(ei, e, E, src, dst);
  float v = lrelu(aS[src] + aD[dst]);
  float x = expf(v - fdec(mx[dst]));
  ex[e] = x;
  atomicAdd(&s[dst], x);
}

__global__ void edge_aggr2(const long long* __restrict__ ei,
                           const float* __restrict__ ex, const float* __restrict__ s,
                           const float* __restrict__ z2, float* __restrict__ acc,
                           int E, int N) {
  int i = blockIdx.x * blockDim.x + threadIdx.x;
  if (i >= (E + N) * CC) return;
  int e = i / CC, f = i - e * CC;
  int src, dst;
  edge_sd(ei, e, E, src, dst);
  float alpha = ex[e] / (s[dst] + SEPS);
  atomicAdd(&acc[(size_t)dst * CC + f], alpha * z2[(size_t)src * CC + f]);
}

// ------------------------------------------- final bias + log_softmax
__global__ void logsoftmax_k(const float* __restrict__ acc,
                             const float* __restrict__ b2,
                             float* __restrict__ out, int N) {
  int n = blockIdx.x * blockDim.x + threadIdx.x;
  if (n >= N) return;
  float v[CC];
  float mx = -INFINITY;
#pragma unroll
  for (int c = 0; c < CC; c++) {
    v[c] = acc[(size_t)n * CC + c] + b2[c];
    mx = fmaxf(mx, v[c]);
  }
  float sum = 0.0f;
#pragma unroll
  for (int c = 0; c < CC; c++) sum += expf(v[c] - mx);
  float lse = mx + logf(sum);
#pragma unroll
  for (int c = 0; c < CC; c++) out[(size_t)n * CC + c] = v[c] - lse;
}

// =====================================================================
extern "C" void kernel_launch(void* const* d_in, const int* in_sizes, int n_in,
                              void* d_out, int out_size, void* d_ws, size_t ws_size,
                              hipStream_t stream) {
  const float*     x     = (const float*)d_in[0];
  const long long* ei    = (const long long*)d_in[1];   // int64 [2,E]
  const float*     W1    = (const float*)d_in[2];
  const float*     attS1 = (const float*)d_in[3];
  const float*     attD1 = (const float*)d_in[4];
  const float*     b1    = (const float*)d_in[5];
  const float*     W2    = (const float*)d_in[6];
  const float*     attS2 = (const float*)d_in[7];
  const float*     attD2 = (const float*)d_in[8];
  const float*     b2    = (const float*)d_in[9];
  float*           out   = (float*)d_out;

  const int N  = in_sizes[0] / DIN;   // 50000
  const int E  = in_sizes[1] / 2;     // 800000
  const int ET = E + N;               // with self-loops

  // ---------------- workspace layout (floats) ----------------
  float* W = (float*)d_ws;
  size_t o = 0;
  float*    acc1 = W + o;              o += (size_t)N * HF;   // zeroed
  float*    s1   = W + o;              o += (size_t)N * NH;   // zeroed
  unsigned* mx1  = (unsigned*)(W + o); o += (size_t)N * NH;   // zeroed
  const long long nzero1 = (long long)o;
  float*    h1   = W + o;              o += (size_t)N * HF;
  float*    ex1  = W + o;              o += (size_t)ET * NH;
  float*    aS1  = W + o;              o += (size_t)N * NH;
  float*    aD1  = W + o;              o += (size_t)N * NH;
  // layer-2 buffers alias dead h1 region (h1 unused after edge_aggr1)
  float*    z2   = h1;                              // N*CC
  float*    acc2 = h1 + (size_t)N * CC;             // N*CC  (zeroed)
  float*    s2   = h1 + 2 * (size_t)N * CC;         // N     (zeroed)
  unsigned* mx2  = (unsigned*)(h1 + 2 * (size_t)N * CC + N); // N (zeroed)
  float*    aS2  = h1 + 2 * (size_t)N * CC + 2 * (size_t)N;  // N
  float*    aD2  = aS2 + N;                                  // N
  float*    ex2  = aD2 + N;                                  // ET

  // ---------------- layer 1 ----------------
  fill_zero_k<<<(unsigned)((nzero1 + 255) / 256), 256, 0, stream>>>(W, nzero1);

  dim3 g1((N + 15) / 16, 2);
  gemm1_wmma<<<g1, 256, 0, stream>>>(x, W1, h1, N);

  int tA = N * NH;
  att1_k<<<(tA + 255) / 256, 256, 0, stream>>>(h1, attS1, attD1, aS1, aD1, N);

  int tE = ET * NH;
  edge_max1<<<(tE + 255) / 256, 256, 0, stream>>>(ei, aS1, aD1, mx1, E, N);
  edge_sum1<<<(tE + 255) / 256, 256, 0, stream>>>(ei, aS1, aD1, mx1, ex1, s1, E, N);
  edge_aggr1<<<(ET + 3) / 4, 256, 0, stream>>>(ei, ex1, s1, h1, acc1, E, N);

  long long tH = (long long)N * HF;
  elu_bias_k<<<(unsigned)((tH + 255) / 256), 256, 0, stream>>>(acc1, b1, tH);

  // ---------------- layer 2 ----------------
  long long nzero2 = (long long)N * CC + 2 * (long long)N;  // acc2,s2,mx2
  fill_zero_k<<<(unsigned)((nzero2 + 255) / 256), 256, 0, stream>>>(acc2, nzero2);

  int nWaves = ((N + 15) / 16) * 3;
  gemm2_wmma<<<(nWaves + 7) / 8, 256, 0, stream>>>(acc1, W2, z2, N, nWaves);

  att2_k<<<(N + 255) / 256, 256, 0, stream>>>(z2, attS2, attD2, aS2, aD2, N);

  edge_max2<<<(ET + 255) / 256, 256, 0, stream>>>(ei, aS2, aD2, mx2, E, N);
  edge_sum2<<<(ET + 255) / 256, 256, 0, stream>>>(ei, aS2, aD2, mx2, ex2, s2, E, N);

  int tF = ET * CC;
  edge_aggr2<<<(tF + 255) / 256, 256, 0, stream>>>(ei, ex2, s2, z2, acc2, E, N);

  logsoftmax_k<<<(N + 255) / 256, 256, 0, stream>>>(acc2, b2, out, N);
}